// ShootingLayer_84679575208171
// MI455X (gfx1250) — compile-verified
//
#include <hip/hip_runtime.h>
#include <math.h>

typedef float v2f __attribute__((ext_vector_type(2)));
typedef float v8f __attribute__((ext_vector_type(8)));

// D = A(16x4) * B(4x16) + C  on the CDNA5 matrix pipe
#define WMMA4(A, B, Cc) \
  __builtin_amdgcn_wmma_f32_16x16x4_f32(false, (A), false, (B), (short)0, (Cc), false, false)

namespace {
constexpr int   Cn    = 8192;
constexpr int   TILES = Cn / 16;     // 512 j-tiles
constexpr int   SPLIT = 8;           // j-strips per i-block (parallelism)
constexpr int   TPW   = TILES / SPLIT; // 64 tiles per wave
constexpr float DT    = 0.25f;       // (T1-T0)/(steps-1)
constexpr int   STATE = 2 * Cn * 3;  // 49152 floats per [2,C,3] state

// workspace layout (float offsets)
constexpr size_t OFF_CS = 0;                              // stage c      [C][3]
constexpr size_t OFF_MS = OFF_CS + (size_t)Cn * 3;        // stage m      [C][3]
constexpr size_t OFF_X2 = OFF_MS + (size_t)Cn * 3;        // |c|^2        [C]
constexpr size_t OFF_AG = OFF_X2 + (size_t)Cn;            // gram A-op    [TILES][32][2]
constexpr size_t OFF_AF = OFF_AG + (size_t)TILES * 64;    // feature A-op [TILES][32][8]
constexpr size_t OFF_KC = OFF_AF + (size_t)TILES * 256;   // k current    [2][C][3]
constexpr size_t OFF_KA = OFF_KC + (size_t)STATE;         // RK4 accum    [2][C][3]
constexpr size_t OFF_PT = OFF_KA + (size_t)STATE;         // strip partials [SPLIT][2][C][3]
}

// ---- write t_span and st[0] = [control_points, momentum] -------------------
__global__ void init_kernel(const float* __restrict__ mom,
                            const float* __restrict__ cp,
                            float* __restrict__ out) {
  int idx = blockIdx.x * blockDim.x + threadIdx.x;
  if (idx < 5) out[idx] = DT * (float)idx;           // linspace(0,1,5)
  int e = idx - 5;
  if (e >= 0 && e < STATE) out[5 + e] = (e < 3 * Cn) ? cp[e] : mom[e - 3 * Cn];
}

// ---- build stage state s + a*k and WMMA operand arrays ---------------------
__global__ void prep_kernel(const float* __restrict__ sb,
                            const float* __restrict__ kp,
                            float a, float* __restrict__ ws) {
  int j = blockIdx.x * blockDim.x + threadIdx.x;
  if (j >= Cn) return;
  float* c_s = ws + OFF_CS;  float* m_s = ws + OFF_MS;
  float* x2  = ws + OFF_X2;  float* Ag  = ws + OFF_AG;  float* Af = ws + OFF_AF;

  float cx = sb[j*3+0] + a * kp[j*3+0];
  float cy = sb[j*3+1] + a * kp[j*3+1];
  float cz = sb[j*3+2] + a * kp[j*3+2];
  float mx = sb[3*Cn + j*3+0] + a * kp[3*Cn + j*3+0];
  float my = sb[3*Cn + j*3+1] + a * kp[3*Cn + j*3+1];
  float mz = sb[3*Cn + j*3+2] + a * kp[3*Cn + j*3+2];
  float q  = cx*cx + cy*cy + cz*cz;

  c_s[j*3+0]=cx; c_s[j*3+1]=cy; c_s[j*3+2]=cz;
  m_s[j*3+0]=mx; m_s[j*3+1]=my; m_s[j*3+2]=mz;
  x2[j] = q;

  int t = j >> 4, jl = j & 15, hi = jl >> 3, v = jl & 7;
  // Gram A-operand (f32 16x4 A-layout): lanes<16 hold K0,K1; lanes>=16 hold K2,K3
  Ag[((size_t)t*32 + jl     )*2 + 0] = 2.f * cx;   // K0
  Ag[((size_t)t*32 + jl     )*2 + 1] = 2.f * cy;   // K1
  Ag[((size_t)t*32 + jl + 16)*2 + 0] = 2.f * cz;   // K2
  Ag[((size_t)t*32 + jl + 16)*2 + 1] = -q;         // K3  -> G = 2 c_j.c_i - |c_j|^2
  // Feature matrix B[j][16] transposed into A-operand chunks; row-in-tile jl = (2k+r) + 8*hi
  float B[16] = {mx, my, mz,
                 mx*cx, mx*cy, mx*cz,
                 my*cx, my*cy, my*cz,
                 mz*cx, mz*cy, mz*cz,
                 0.f, 0.f, 0.f, 0.f};
#pragma unroll
  for (int f = 0; f < 16; ++f)
    Af[((size_t)t*32 + hi*16 + f)*8 + v] = B[f];
}

// ---- fused pairwise kernel: K on the fly + Acc^T = B^T @ K'^T --------------
__global__ __launch_bounds__(128) void pair_kernel(float* __restrict__ ws) {
  const float* c_s = ws + OFF_CS;
  const float* m_s = ws + OFF_MS;
  const float* x2  = ws + OFF_X2;
  const float* Ag  = ws + OFF_AG;
  const float* Af  = ws + OFF_AF;
  float*       prt = ws + OFF_PT;

  int lane = threadIdx.x & 31;
  int wv   = threadIdx.x >> 5;
  int wid  = blockIdx.x * 4 + wv;
  int iblk = wid / SPLIT, strip = wid % SPLIT;
  int col  = lane & 15,  hi = lane >> 4;
  int i    = iblk * 16 + col;

  // Gram B-operand (f32 4x16 B-layout): lanes<16: K0,K1 = (cx,cy); lanes>=16: K2,K3 = (cz,1)
  v2f bg;
  bg.x = hi ? c_s[i*3+2] : c_s[i*3+0];
  bg.y = hi ? 1.0f       : c_s[i*3+1];

  v8f zf = {0,0,0,0,0,0,0,0};
  v8f acc0 = zf, acc1 = zf;

  int t0 = strip * TPW;
  for (int t = t0; t < t0 + TPW; ++t) {
    v2f ag = *(const v2f*)(Ag + ((size_t)t*32 + lane)*2);
    v8f g  = WMMA4(ag, bg, zf);                       // G(j,i) = 2 c_j.c_i - |c_j|^2
    v8f af = *(const v8f*)(Af + ((size_t)t*32 + lane)*8);
    float k0=__expf(g[0]), k1=__expf(g[1]), k2=__expf(g[2]), k3=__expf(g[3]);
    float k4=__expf(g[4]), k5=__expf(g[5]), k6=__expf(g[6]), k7=__expf(g[7]);
    // C/D-layout VGPR pair (v,v+1) is a valid f32 B-operand over K-rows {v,v+1,v+8,v+9}
    v2f a0; a0.x=af[0]; a0.y=af[1];  v2f b0; b0.x=k0; b0.y=k1;
    v2f a1; a1.x=af[2]; a1.y=af[3];  v2f b1; b1.x=k2; b1.y=k3;
    v2f a2; a2.x=af[4]; a2.y=af[5];  v2f b2; b2.x=k4; b2.y=k5;
    v2f a3; a3.x=af[6]; a3.y=af[7];  v2f b3; b3.x=k6; b3.y=k7;
    acc0 = WMMA4(a0, b0, acc0);      // two independent accumulators break the
    acc1 = WMMA4(a1, b1, acc1);      // WMMA->WMMA RAW chain for pipelining
    acc0 = WMMA4(a2, b2, acc0);
    acc1 = WMMA4(a3, b3, acc1);
  }

  float ex = __expf(-x2[i]);                          // factored e^{-|c_i|^2}
  __shared__ float smem[4][256];
  float* sm = smem[wv];
#pragma unroll
  for (int v = 0; v < 8; ++v)
    sm[(v + 8*hi)*16 + col] = (acc0[v] + acc1[v]) * ex;
  __syncthreads();

  if (lane < 16) {
    float F[12];
#pragma unroll
    for (int f = 0; f < 12; ++f) F[f] = sm[f*16 + col];
    float mx=m_s[i*3+0], my=m_s[i*3+1], mz=m_s[i*3+2];
    float cx=c_s[i*3+0], cy=c_s[i*3+1], cz=c_s[i*3+2];
    float s2  = mx*F[0] + my*F[1] + mz*F[2];          // Σ K (m_i.m_j)
    float s3x = mx*F[3] + my*F[6] + mz*F[9];
    float s3y = mx*F[4] + my*F[7] + mz*F[10];
    float s3z = mx*F[5] + my*F[8] + mz*F[11];
    float* p = prt + (size_t)strip * STATE;
    p[i*3+0] = F[0]; p[i*3+1] = F[1]; p[i*3+2] = F[2];          // dc = K@m
    p[3*Cn + i*3+0] = 2.f*(cx*s2 - s3x);                        // dm
    p[3*Cn + i*3+1] = 2.f*(cy*s2 - s3y);
    p[3*Cn + i*3+2] = 2.f*(cz*s2 - s3z);
  }
}

// ---- deterministic strip reduction + RK4 weight accumulation ---------------
__global__ void reduce_kernel(float* __restrict__ ws, float w, int first) {
  int e = blockIdx.x * blockDim.x + threadIdx.x;
  if (e >= STATE) return;
  const float* prt = ws + OFF_PT;
  float s = 0.f;
#pragma unroll
  for (int q = 0; q < SPLIT; ++q) s += prt[(size_t)q * STATE + e];
  ws[OFF_KC + e] = s;
  float* ka = ws + OFF_KA;
  ka[e] = first ? w * s : ka[e] + w * s;
}

// ---- s_next = s + dt/6 * (k1 + 2k2 + 2k3 + k4) -----------------------------
__global__ void advance_kernel(const float* __restrict__ sb,
                               const float* __restrict__ ws,
                               float* __restrict__ sn) {
  int e = blockIdx.x * blockDim.x + threadIdx.x;
  if (e >= STATE) return;
  sn[e] = sb[e] + (DT / 6.f) * ws[OFF_KA + e];
}

extern "C" void kernel_launch(void* const* d_in, const int* in_sizes, int n_in,
                              void* d_out, int out_size, void* d_ws, size_t ws_size,
                              hipStream_t stream) {
  (void)in_sizes; (void)n_in; (void)out_size; (void)ws_size;
  const float* mom = (const float*)d_in[0];   // momentum       [C,3]
  const float* cp  = (const float*)d_in[1];   // control_points [C,3]
  float* out = (float*)d_out;                 // [5] t_span | [5,2,C,3] st
  float* ws  = (float*)d_ws;

  init_kernel<<<(5 + STATE + 255) / 256, 256, 0, stream>>>(mom, cp, out);

  const float avals[4] = {0.f, 0.5f * DT, 0.5f * DT, DT};
  const float wvals[4] = {1.f, 2.f, 2.f, 1.f};

  for (int s = 0; s < 4; ++s) {
    const float* sb = out + 5 + (size_t)s * STATE;
    float*       sn = out + 5 + (size_t)(s + 1) * STATE;
    for (int st = 0; st < 4; ++st) {
      const float* kp = (st == 0) ? sb : (ws + OFF_KC);
      prep_kernel<<<Cn / 256, 256, 0, stream>>>(sb, kp, avals[st], ws);
      pair_kernel<<<(TILES * SPLIT) / 4, 128, 0, stream>>>(ws);
      reduce_kernel<<<STATE / 256, 256, 0, stream>>>(ws, wvals[st], st == 0);
    }
    advance_kernel<<<STATE / 256, 256, 0, stream>>>(sb, ws, sn);
  }
}